// CouchesintermediairesGNN_90082644066768
// MI455X (gfx1250) — compile-verified
//
#include <hip/hip_runtime.h>
#include <hip/hip_bf16.h>
#include <math.h>

typedef __attribute__((ext_vector_type(2))) float v2f;
typedef __attribute__((ext_vector_type(8))) float v8f;

#define THREADS 256

// ---------------------------------------------------------------------------
// Guaranteed-hardware f32 atomic add (no return): GLOBAL_ATOMIC_ADD_F32,
// tracked by STOREcnt, drained by implicit wait-idle at s_endpgm.
// ---------------------------------------------------------------------------
__device__ __forceinline__ void hw_atomic_add_f32(float* p, float v) {
    asm volatile("global_atomic_add_f32 %0, %1, off"
                 :
                 : "v"(p), "v"(v)
                 : "memory");
}

// ---------------------------------------------------------------------------
// Zero sum_w (workspace) and the sum_features region of the output.
// ---------------------------------------------------------------------------
__global__ void gnn_zero_kernel(float* __restrict__ sumw, float* __restrict__ out, int N) {
    int i = blockIdx.x * blockDim.x + threadIdx.x;
    int total = N * 20;
    if (i < total) {
        sumw[i] = 0.0f;
        int n = i / 20;
        int c = i - n * 20;
        out[(size_t)n * 40 + 20 + c] = 0.0f;   // out[n,1,c] = sum_features accumulator
    }
}

// ---------------------------------------------------------------------------
// Edge MLP: Linear(1,64) -> ReLU -> Linear(64,10). Pure function of scalar d.
// ~704 FMA, all weight loads are wave-uniform (SMEM path).
// ---------------------------------------------------------------------------
__device__ __forceinline__ void edge_mlp(float d,
                                         const float* __restrict__ w1,
                                         const float* __restrict__ b1,
                                         const float* __restrict__ w2,
                                         const float* __restrict__ b2,
                                         float mlp[10]) {
#pragma unroll
    for (int j = 0; j < 10; ++j) mlp[j] = b2[j];
    for (int k = 0; k < 64; ++k) {
        float h = fmaf(d, w1[k], b1[k]);
        h = fmaxf(h, 0.0f);
#pragma unroll
        for (int j = 0; j < 10; ++j) mlp[j] = fmaf(h, w2[k * 10 + j], mlp[j]);
    }
}

__device__ __forceinline__ int bucket_of(float d) {
    int idx = (int)(d * 10.0f);       // trunc toward zero, matches astype(int32)
    idx = idx < 0 ? 0 : idx;
    idx = idx > 9 ? 9 : idx;
    return idx;
}

// ---------------------------------------------------------------------------
// Pass 1: per-edge eac -> atomic segment sum over src. One-hot contributes a
// single 1.0 (structural zeros skipped => 11 atomics/edge instead of 20).
// ---------------------------------------------------------------------------
__global__ void gnn_pass1_kernel(const int* __restrict__ src,
                                 const float* __restrict__ attr,
                                 const float* __restrict__ w1, const float* __restrict__ b1,
                                 const float* __restrict__ w2, const float* __restrict__ b2,
                                 float* __restrict__ sumw, int E) {
    int e = blockIdx.x * blockDim.x + threadIdx.x;
    if (e >= E) return;
    float d = attr[e];
    int   s = src[e];
    float mlp[10];
    edge_mlp(d, w1, b1, w2, b2, mlp);
    float* base = sumw + (size_t)s * 20;
    hw_atomic_add_f32(base + bucket_of(d), 1.0f);
#pragma unroll
    for (int j = 0; j < 10; ++j) hw_atomic_add_f32(base + 10 + j, mlp[j]);
}

// ---------------------------------------------------------------------------
// Pass 2: recompute eac, normalize by sum_w[src], weight by rho, atomic
// scatter into out[:,1,:] (sum_features).
// ---------------------------------------------------------------------------
__global__ void gnn_pass2_kernel(const int* __restrict__ src,
                                 const int* __restrict__ dst,
                                 const float* __restrict__ attr,
                                 const float* __restrict__ x,
                                 const float* __restrict__ a_p,
                                 const float* __restrict__ b_p,
                                 const float* __restrict__ w1, const float* __restrict__ b1,
                                 const float* __restrict__ w2, const float* __restrict__ b2,
                                 const float* __restrict__ sumw,
                                 float* __restrict__ out, int E) {
    int e = blockIdx.x * blockDim.x + threadIdx.x;
    if (e >= E) return;
    float d = attr[e];
    int   s = src[e];
    int   t = dst[e];
    float aa = a_p[0];
    float bb = b_p[0];
    float ca = 1.0f - aa;

    float mlp[10];
    edge_mlp(d, w1, b1, w2, b2, mlp);
    int idx = bucket_of(d);

    float sw[20], hj[20], hp[20];
    {
        const float4* p0 = reinterpret_cast<const float4*>(sumw + (size_t)s * 20);
        const float4* p1 = reinterpret_cast<const float4*>(x + (size_t)s * 40);
        const float4* p2 = reinterpret_cast<const float4*>(x + (size_t)t * 40);
#pragma unroll
        for (int q = 0; q < 5; ++q) {
            float4 v0 = p0[q], v1 = p1[q], v2 = p2[q];
            sw[4*q+0] = v0.x; sw[4*q+1] = v0.y; sw[4*q+2] = v0.z; sw[4*q+3] = v0.w;
            hj[4*q+0] = v1.x; hj[4*q+1] = v1.y; hj[4*q+2] = v1.z; hj[4*q+3] = v1.w;
            hp[4*q+0] = v2.x; hp[4*q+1] = v2.y; hp[4*q+2] = v2.z; hp[4*q+3] = v2.w;
        }
    }

    float* obase = out + (size_t)s * 40 + 20;  // sum_features row of src
#pragma unroll
    for (int c = 0; c < 20; ++c) {
        float eac = (c < 10) ? ((c == idx) ? 1.0f : 0.0f) : mlp[c - 10];
        float swc = sw[c];
        float wt  = (swc != 0.0f) ? (eac / swc) : 0.01f;
        float mag = fabsf(aa * hj[c] - ca * hp[c]);
        float rho = powf(mag, bb);
        hw_atomic_add_f32(obase + c, rho * wt);
    }
}

// ---------------------------------------------------------------------------
// Node update with f32 WMMA (V_WMMA_F32_16X16X4_F32, exact f32 math):
//   out0 = sigmoid([x0 | sf] @ [g1;g2]^T + bias)
// One wave per 16-node tile. K = 40 (10 steps of 4). Two N tiles (cols 0-15,
// 16-19 with zero-padded B columns 20-31).
//
// Fragment layouts per CDNA5 ISA 7.12.2:
//   A 16x4 f32 : lanes 0-15 hold M=lane  {K=k0, k0+1};  lanes 16-31 {K=k0+2, k0+3}
//   B 4x16 f32 : reg0 = row k0, reg1 = row k0+1, with k0 += 2 for lanes 16-31
//   C 16x16 f32: reg r = row (r + 8*(lane/16)), col = lane%16
// ---------------------------------------------------------------------------
__global__ void gnn_node_wmma_kernel(const float* __restrict__ x,
                                     const float* __restrict__ g1,
                                     const float* __restrict__ g2,
                                     const float* __restrict__ bias,
                                     float* __restrict__ out, int nTiles) {
    int gid  = blockIdx.x * blockDim.x + threadIdx.x;
    int wave = gid >> 5;               // one wave32 per 16-node tile
    int lane = threadIdx.x & 31;
    if (wave >= nTiles) return;        // uniform per wave: EXEC stays all-1s
    int laneLo = lane & 15;
    int laneHi = lane >> 4;

    // B fragments: G[k][j] = (k<20 ? gamma1[j][k] : gamma2[j][k-20])
    v2f bf0[10], bf1[10];
#pragma unroll
    for (int kk = 0; kk < 10; ++kk) {
        int k0 = kk * 4 + laneHi * 2;
        int col0 = laneLo;             // tile 0: cols 0..15, always valid
        v2f b0;
        b0.x = (k0     < 20) ? g1[col0 * 20 + k0]     : g2[col0 * 20 + k0 - 20];
        b0.y = (k0 + 1 < 20) ? g1[col0 * 20 + k0 + 1] : g2[col0 * 20 + k0 - 19];
        bf0[kk] = b0;
        int col1 = 16 + laneLo;        // tile 1: cols 16..31, valid only < 20
        v2f b1v; b1v.x = 0.0f; b1v.y = 0.0f;
        if (col1 < 20) {
            b1v.x = (k0     < 20) ? g1[col1 * 20 + k0]     : g2[col1 * 20 + k0 - 20];
            b1v.y = (k0 + 1 < 20) ? g1[col1 * 20 + k0 + 1] : g2[col1 * 20 + k0 - 19];
        }
        bf1[kk] = b1v;
    }

    // A fragments: row m = [ x[m,0,0..19] | sum_features[m,0..19] ]
    int m = wave * 16 + laneLo;
    const float* xr = x   + (size_t)m * 40;        // k in [0,20)
    const float* sr = out + (size_t)m * 40 + 20;   // k in [20,40) -> sf
    v2f af[10];
#pragma unroll
    for (int kk = 0; kk < 10; ++kk) {
        int k0 = kk * 4 + laneHi * 2;
        v2f a;
        if (k0 < 20) { a.x = xr[k0];      a.y = xr[k0 + 1];  }
        else         { a.x = sr[k0 - 20]; a.y = sr[k0 - 19]; }
        af[kk] = a;
    }

    v8f c0 = {0,0,0,0,0,0,0,0};
    v8f c1 = {0,0,0,0,0,0,0,0};
#pragma unroll
    for (int kk = 0; kk < 10; ++kk) {
        c0 = __builtin_amdgcn_wmma_f32_16x16x4_f32(false, af[kk], false, bf0[kk],
                                                   (short)0, c0, false, false);
        c1 = __builtin_amdgcn_wmma_f32_16x16x4_f32(false, af[kk], false, bf1[kk],
                                                   (short)0, c1, false, false);
    }

    float bias0 = bias[laneLo];
    float bias1 = (laneLo < 4) ? bias[16 + laneLo] : 0.0f;
#pragma unroll
    for (int r = 0; r < 8; ++r) {
        int node = wave * 16 + r + 8 * laneHi;
        float z0 = c0[r] + bias0;
        out[(size_t)node * 40 + laneLo] = 1.0f / (1.0f + expf(-z0));
        if (laneLo < 4) {
            float z1 = c1[r] + bias1;
            out[(size_t)node * 40 + 16 + laneLo] = 1.0f / (1.0f + expf(-z1));
        }
    }
}

// Scalar fallback for a possible N % 16 tail (N = 100k is exactly divisible).
__global__ void gnn_node_tail_kernel(const float* __restrict__ x,
                                     const float* __restrict__ g1,
                                     const float* __restrict__ g2,
                                     const float* __restrict__ bias,
                                     float* __restrict__ out, int start, int N) {
    int n = start + blockIdx.x * blockDim.x + threadIdx.x;
    if (n >= N) return;
    const float* xr = x   + (size_t)n * 40;
    const float* sr = out + (size_t)n * 40 + 20;
    for (int j = 0; j < 20; ++j) {
        float acc = bias[j];
        for (int h = 0; h < 20; ++h)
            acc += xr[h] * g1[j * 20 + h] + sr[h] * g2[j * 20 + h];
        out[(size_t)n * 40 + j] = 1.0f / (1.0f + expf(-acc));
    }
}

// ---------------------------------------------------------------------------
extern "C" void kernel_launch(void* const* d_in, const int* in_sizes, int n_in,
                              void* d_out, int out_size, void* d_ws, size_t ws_size,
                              hipStream_t stream) {
    const float* x    = (const float*)d_in[0];   // [N,2,20]
    const int*   ei   = (const int*)  d_in[1];   // [2,E]
    const float* attr = (const float*)d_in[2];   // [E,1]
    const float* a_p  = (const float*)d_in[3];   // [1]
    const float* b_p  = (const float*)d_in[4];   // [1]
    const float* g1   = (const float*)d_in[5];   // [20,20]
    const float* g2   = (const float*)d_in[6];   // [20,20]
    const float* bias = (const float*)d_in[7];   // [20]
    const float* W1   = (const float*)d_in[8];   // [1,64]
    const float* b1   = (const float*)d_in[9];   // [64]
    const float* W2   = (const float*)d_in[10];  // [64,10]
    const float* b2   = (const float*)d_in[11];  // [10]
    float* out = (float*)d_out;                  // [N,2,20]

    int N = in_sizes[0] / 40;
    int E = in_sizes[1] / 2;
    const int* src = ei;
    const int* dst = ei + E;
    float* sumw = (float*)d_ws;                  // [N,20] segment sums (8 MB)

    int zblocks = (N * 20 + THREADS - 1) / THREADS;
    gnn_zero_kernel<<<zblocks, THREADS, 0, stream>>>(sumw, out, N);

    int eblocks = (E + THREADS - 1) / THREADS;
    gnn_pass1_kernel<<<eblocks, THREADS, 0, stream>>>(src, attr, W1, b1, W2, b2, sumw, E);
    gnn_pass2_kernel<<<eblocks, THREADS, 0, stream>>>(src, dst, attr, x, a_p, b_p,
                                                      W1, b1, W2, b2, sumw, out, E);

    int nTiles = N / 16;
    if (nTiles > 0) {
        int wBlocks = (nTiles * 32 + THREADS - 1) / THREADS;  // 1 wave per tile
        gnn_node_wmma_kernel<<<wBlocks, THREADS, 0, stream>>>(x, g1, g2, bias, out, nTiles);
    }
    int rem = N - nTiles * 16;
    if (rem > 0) {
        gnn_node_tail_kernel<<<(rem + THREADS - 1) / THREADS, THREADS, 0, stream>>>(
            x, g1, g2, bias, out, nTiles * 16, N);
    }
}